// CapsuleLinear_1460288880649
// MI455X (gfx1250) — compile-verified
//
#include <hip/hip_runtime.h>

// Problem constants (must match reference)
#define N_TOT    64
#define IN_CAPS  256
#define OUT_CAPS 128
#define IN_LEN   16
#define OUT_LEN  32
#define NT       8          // n-rows per workgroup tile
#define NITER    3

// LDS layout (dwords). P block per in-cap padded 256->260 dwords: multiple of
// 4 dwords (keeps 16B alignment for b128 LDS loads) and 260 mod 64 = 4 so the
// i-parallel dot pass sees at most 2-way bank conflicts.
#define PI_STRIDE 260
#define C_OFF  (IN_CAPS * PI_STRIDE)               // coupling coeffs c[n][i]
#define B_OFF  (C_OFF + NT * IN_CAPS)              // logits b[n][i]
#define V_OFF  (B_OFF + NT * IN_CAPS)              // v[n][l]
#define LDS_DW (V_OFF + NT * OUT_LEN)              // = 70912 dwords = 283648 B

typedef __attribute__((ext_vector_type(2))) float v2f;
typedef __attribute__((ext_vector_type(8))) float v8f;

__global__ void __launch_bounds__(256)
capsule_routing_kernel(const float* __restrict__ x,     // [N, I, 16]
                       const float* __restrict__ w,     // [O, I, 16, 32]
                       float* __restrict__ out)         // [N, O, 32]
{
    extern __shared__ float lds[];
    const int tid  = threadIdx.x;
    const int lane = tid & 31;
    const int wave = tid >> 5;          // 8 waves
    const int o    = blockIdx.y;        // out-capsule
    const int n0   = blockIdx.x * NT;   // n-tile base

    // ---------------- Phase 1: priors tile via V_WMMA_F32_16X16X4_F32 -----
    // Each wave computes priors[n0:n0+8, i, 0:32] for its 32 in-caps.
    {
        const int  M        = lane & 15;        // A-matrix row (n within tile)
        const int  hi       = lane >> 4;        // lane half selects K pair
        const bool rowValid = (M < NT);
        // Clamp invalid rows to row 0: loads stay in-bounds and unconditional,
        // the zeroing happens via v_cndmask selects (no exec-mask branches).
        const float* xbase  = x + (size_t)(n0 + (rowValid ? M : 0)) * IN_CAPS * IN_LEN;

        for (int ii = 0; ii < 32; ++ii) {
            const int i = wave * 32 + ii;
            const float* xr    = xbase + (size_t)i * IN_LEN + 2 * hi;
            const float* wbase = w + (((size_t)o * IN_CAPS + i) * IN_LEN) * OUT_LEN + (lane & 15);

            // A fragments for all 4 K-steps: one clause of 4x b64 loads.
            float2 q[4];
#pragma unroll
            for (int s = 0; s < 4; ++s)
                q[s] = *reinterpret_cast<const float2*>(xr + 4 * s);

            // Speculative prefetch of next in-cap's weight slab (OOB-safe).
            __builtin_prefetch(wbase + IN_LEN * OUT_LEN, 0, 3);

            v8f acc0 = {};   // l = 0..15
            v8f acc1 = {};   // l = 16..31
#pragma unroll
            for (int s = 0; s < 4; ++s) {
                const int k0 = 4 * s + 2 * hi;  // ISA 16x4 fp32 A layout
                v2f a = { rowValid ? q[s].x : 0.0f,
                          rowValid ? q[s].y : 0.0f };
                const float* wp = wbase + (size_t)k0 * OUT_LEN;
                v2f b0 = { wp[0],  wp[OUT_LEN]      };   // cols 0..15
                v2f b1 = { wp[16], wp[OUT_LEN + 16] };   // cols 16..31
                acc0 = __builtin_amdgcn_wmma_f32_16x16x4_f32(
                           false, a, false, b0, (short)0, acc0, false, false);
                acc1 = __builtin_amdgcn_wmma_f32_16x16x4_f32(
                           false, a, false, b1, (short)0, acc1, false, false);
            }

            // C/D layout: lanes 0-15 hold N=lane, VGPR r = row M=r (rows 8-15
            // live in lanes 16-31 and are exact zeros here -> discard).
            if (lane < 16) {
#pragma unroll
                for (int r = 0; r < NT; ++r) {
                    lds[i * PI_STRIDE + r * OUT_LEN + lane]      = acc0[r];
                    lds[i * PI_STRIDE + r * OUT_LEN + 16 + lane] = acc1[r];
                }
            }
        }
    }

    // ---------------- Phase 2: dynamic routing in LDS ----------------------
    float* cb = lds + C_OFF;
    float* bb = lds + B_OFF;
    float* vb = lds + V_OFF;
    const int n = wave;        // wave w owns tile row n = w
    const int l = lane;        // lane = out_len element

#pragma unroll
    for (int t = 0; t < 8; ++t) bb[n * IN_CAPS + lane + 32 * t] = 0.0f;
    __syncthreads();           // priors + logits visible to all

    for (int it = 0; it < NITER; ++it) {
        // softmax over in_caps for row n (one wave per row, shuffle reductions)
        float bv[8], ev[8];
        float m = -3.4e38f;
#pragma unroll
        for (int t = 0; t < 8; ++t) {
            bv[t] = bb[n * IN_CAPS + lane + 32 * t];
            m = fmaxf(m, bv[t]);
        }
#pragma unroll
        for (int off = 16; off > 0; off >>= 1) m = fmaxf(m, __shfl_xor(m, off, 32));
        float z = 0.0f;
#pragma unroll
        for (int t = 0; t < 8; ++t) { ev[t] = __expf(bv[t] - m); z += ev[t]; }
#pragma unroll
        for (int off = 16; off > 0; off >>= 1) z += __shfl_xor(z, off, 32);
        const float rz = 1.0f / z;
#pragma unroll
        for (int t = 0; t < 8; ++t) cb[n * IN_CAPS + lane + 32 * t] = ev[t] * rz;
        __syncthreads();

        // s[n,l] = sum_i c[n,i] * P[i,n,l]
        // c read as float4 broadcasts (ds_load_b128), P lane-parallel over l.
        float s = 0.0f;
#pragma unroll 4
        for (int i = 0; i < IN_CAPS; i += 4) {
            const float4 cv = *reinterpret_cast<const float4*>(cb + n * IN_CAPS + i);
            s += cv.x * lds[(i + 0) * PI_STRIDE + n * OUT_LEN + l];
            s += cv.y * lds[(i + 1) * PI_STRIDE + n * OUT_LEN + l];
            s += cv.z * lds[(i + 2) * PI_STRIDE + n * OUT_LEN + l];
            s += cv.w * lds[(i + 3) * PI_STRIDE + n * OUT_LEN + l];
        }

        // squash over l (wave butterfly -> all lanes get ||s||^2)
        float sq = s * s;
#pragma unroll
        for (int off = 16; off > 0; off >>= 1) sq += __shfl_xor(sq, off, 32);
        const float scale = (sq / (1.0f + sq)) * rsqrtf(sq);
        const float v = s * scale;

        if (it == NITER - 1) {
            out[(((size_t)(n0 + n)) * OUT_CAPS + o) * OUT_LEN + l] = v;
        } else {
            vb[n * OUT_LEN + l] = v;
            __syncthreads();
            // Preload v[n,0:32] once into registers (8x ds_load_b128).
            float4 vv[8];
#pragma unroll
            for (int q4 = 0; q4 < 8; ++q4)
                vv[q4] = *reinterpret_cast<const float4*>(vb + n * OUT_LEN + 4 * q4);
            // b[n,i] += sum_l P[i,n,l] * v[n,l]; each lane owns 8 in-caps,
            // P rows read as float4 (ds_load_b128, 16B-aligned via PI_STRIDE).
#pragma unroll
            for (int t = 0; t < 8; ++t) {
                const int i = lane + 32 * t;
                const float4* Pp =
                    reinterpret_cast<const float4*>(lds + i * PI_STRIDE + n * OUT_LEN);
                float d = 0.0f;
#pragma unroll
                for (int q4 = 0; q4 < 8; ++q4) {
                    const float4 pv = Pp[q4];
                    d += pv.x * vv[q4].x + pv.y * vv[q4].y
                       + pv.z * vv[q4].z + pv.w * vv[q4].w;
                }
                bb[n * IN_CAPS + i] += d;
            }
            __syncthreads();
        }
    }
}

extern "C" void kernel_launch(void* const* d_in, const int* in_sizes, int n_in,
                              void* d_out, int out_size, void* d_ws, size_t ws_size,
                              hipStream_t stream) {
    const float* x   = (const float*)d_in[0];   // [64, 256, 16]
    const float* w   = (const float*)d_in[1];   // [128, 256, 16, 32]
    float*       out = (float*)d_out;           // [64, 128, 32]

    const size_t lds_bytes = (size_t)LDS_DW * sizeof(float);   // 283,648 B < 320 KB
    (void)hipFuncSetAttribute((const void*)capsule_routing_kernel,
                              hipFuncAttributeMaxDynamicSharedMemorySize,
                              (int)lds_bytes);

    dim3 grid(N_TOT / NT, OUT_CAPS);   // n-tile fastest: blocks sharing weight[o]
                                       // co-scheduled -> weight[o] stays in L2
    capsule_routing_kernel<<<grid, 256, lds_bytes, stream>>>(x, w, out);
}